// HeteroGAT_31842887533275
// MI455X (gfx1250) — compile-verified
//
#include <hip/hip_runtime.h>
#include <math.h>

#define NAUT 50000
#define NPAP 50000
#define FIN  128
#define NEDGE 800000
#define NEG_SLOPE 0.2f

typedef float v2f __attribute__((ext_vector_type(2)));
typedef float v8f __attribute__((ext_vector_type(8)));

__device__ __forceinline__ float lrelu(float v) { return v > 0.f ? v : NEG_SLOPE * v; }

// float atomic max via sign-aware integer atomics (buffer must be init'd to -inf)
__device__ __forceinline__ void atomicMaxF(float* addr, float val) {
  if (val >= 0.f) atomicMax((int*)addr, __float_as_int(val));
  else            atomicMin((unsigned int*)addr, __float_as_uint(val));
}

// ---------------------------------------------------------------------------
// GEMM C[M,N] = A[M,K] @ B[K,N], row-major, f32, via V_WMMA_F32_16X16X4_F32.
// One wave computes a 16x64 strip of C (4 accumulators), reusing its A frag.
// Requires M%16==0, N%64==0, K%4==0 (true for all 8 GEMMs here).
// A frag layout: lane l holds A[m0+(l&15), k + 2*(l>>4) + j] in vgpr j.
// B frag layout: lane l holds B[k + 2*(l>>4) + j, n0+(l&15)] in vgpr j.
// C frag layout: vgpr r, lane l -> C[m0 + r + (l>=16 ? 8 : 0), n0+(l&15)].
// ---------------------------------------------------------------------------
__global__ void gemm_f32_wmma(const float* __restrict__ A, const float* __restrict__ B,
                              float* __restrict__ C, int M, int N, int K) {
  const int lane = threadIdx.x & 31;
  const int wave = (blockIdx.x * blockDim.x + threadIdx.x) >> 5;
  const int n4   = N >> 6;                 // 64-col strips
  const int mt   = wave / n4;
  const int nt   = wave - mt * n4;
  if (mt >= (M >> 4)) return;              // wave-uniform: EXEC stays all-ones
  const int m0   = mt << 4;
  const int n0   = nt << 6;
  const int mrow = m0 + (lane & 15);
  const int kh   = (lane >> 4) << 1;       // 0 or 2
  const int col  = n0 + (lane & 15);

  v8f acc0 = {}, acc1 = {}, acc2 = {}, acc3 = {};
  for (int k = 0; k < K; k += 4) {
    v2f a = *(const v2f*)(A + (size_t)mrow * K + (k + kh));
    const float* Bp = B + (size_t)(k + kh) * N + col;
    v2f b0 = { Bp[0],  Bp[N]      };
    v2f b1 = { Bp[16], Bp[N + 16] };
    v2f b2 = { Bp[32], Bp[N + 32] };
    v2f b3 = { Bp[48], Bp[N + 48] };
    acc0 = __builtin_amdgcn_wmma_f32_16x16x4_f32(false, a, false, b0, (short)0, acc0, false, false);
    acc1 = __builtin_amdgcn_wmma_f32_16x16x4_f32(false, a, false, b1, (short)0, acc1, false, false);
    acc2 = __builtin_amdgcn_wmma_f32_16x16x4_f32(false, a, false, b2, (short)0, acc2, false, false);
    acc3 = __builtin_amdgcn_wmma_f32_16x16x4_f32(false, a, false, b3, (short)0, acc3, false, false);
  }
  const int r0 = m0 + ((lane >> 4) << 3);
  float* Cp = C + (size_t)r0 * N + col;
#pragma unroll
  for (int r = 0; r < 8; ++r) {
    Cp[(size_t)r * N + 0]  = acc0[r];
    Cp[(size_t)r * N + 16] = acc1[r];
    Cp[(size_t)r * N + 32] = acc2[r];
    Cp[(size_t)r * N + 48] = acc3[r];
  }
}

// a[n,h] = sum_c Hf[n,h*C+c] * att[h*C+c]
__global__ void attn_score(const float* __restrict__ Hf, const float* __restrict__ att,
                           float* __restrict__ out, int Nn, int H, int C) {
  int t = blockIdx.x * blockDim.x + threadIdx.x;
  if (t >= Nn * H) return;
  int n = t / H, h = t - n * H;
  const float* hp = Hf + (size_t)n * H * C + (size_t)h * C;
  const float* ap = att + (size_t)h * C;
  float s = 0.f;
  for (int c = 0; c < C; c += 4) {
    float4 hv = *(const float4*)(hp + c);
    float4 av = *(const float4*)(ap + c);
    s += hv.x * av.x + hv.y * av.y + hv.z * av.z + hv.w * av.w;
  }
  out[t] = s;
}

__global__ void fill_f32(float* __restrict__ p, float v, int n) {
  int t = blockIdx.x * blockDim.x + threadIdx.x;
  if (t < n) p[t] = v;
}

// pass 1: segment max of leaky_relu(a_src[src]+a_dst[dst]) per (dst, head)
__global__ void edge_logit_max(const int* __restrict__ src, const int* __restrict__ dst,
                               const float* __restrict__ as_, const float* __restrict__ ad_,
                               float* __restrict__ emax, int E, int H) {
  int t = blockIdx.x * blockDim.x + threadIdx.x;
  if (t >= E * H) return;
  int e = t / H, h = t - e * H;
  int s = src[e], d = dst[e];
  float v = lrelu(as_[s * H + h] + ad_[d * H + h]);
  atomicMaxF(&emax[d * H + h], v);
}

// pass 2: denom[dst,h] += exp(v - emax[dst,h])
__global__ void edge_denom(const int* __restrict__ src, const int* __restrict__ dst,
                           const float* __restrict__ as_, const float* __restrict__ ad_,
                           const float* __restrict__ emax, float* __restrict__ den,
                           int E, int H) {
  int t = blockIdx.x * blockDim.x + threadIdx.x;
  if (t >= E * H) return;
  int e = t / H, h = t - e * H;
  int s = src[e], d = dst[e];
  float v = lrelu(as_[s * H + h] + ad_[d * H + h]);
  float w = __expf(v - emax[d * H + h]);
  atomicAdd(&den[d * H + h], w);
}

// pass 3: out[dst,h,c] += alpha * Hs[src,h,c]. One wave per edge, HC/32 ch/lane.
__global__ void edge_scatter(const int* __restrict__ src, const int* __restrict__ dst,
                             const float* __restrict__ Hs,
                             const float* __restrict__ as_, const float* __restrict__ ad_,
                             const float* __restrict__ emax, const float* __restrict__ den,
                             float* __restrict__ out, int E, int H, int C) {
  int wave = (blockIdx.x * blockDim.x + threadIdx.x) >> 5;
  int lane = threadIdx.x & 31;
  if (wave >= E) return;
  int s = src[wave], d = dst[wave];
  int HC  = H * C;
  int per = HC >> 5;               // 8 (layer 1) or 2 (layer 2), divides C
  int c0  = lane * per;
  int h   = c0 / C;
  float v = lrelu(as_[s * H + h] + ad_[d * H + h]);
  float w = __expf(v - emax[d * H + h]);
  float alpha = w / (den[d * H + h] + 1e-16f);
  const float* hp = Hs  + (size_t)s * HC + c0;
  float*       op = out + (size_t)d * HC + c0;
  for (int i = 0; i < per; ++i) atomicAdd(op + i, alpha * hp[i]);
}

__global__ void bias_add(float* __restrict__ out, const float* __restrict__ b, int n, int HC) {
  int t = blockIdx.x * blockDim.x + threadIdx.x;
  if (t < n) out[t] += b[t & (HC - 1)];   // HC is 256 or 64 (pow2)
}

extern "C" void kernel_launch(void* const* d_in, const int* in_sizes, int n_in,
                              void* d_out, int out_size, void* d_ws, size_t ws_size,
                              hipStream_t stream) {
  const float* x_author = (const float*)d_in[0];
  const float* x_paper  = (const float*)d_in[1];
  const int*   e_ap     = (const int*)d_in[2];   // [2,E]: row0 src(author), row1 dst(paper)
  const int*   e_pa     = (const int*)d_in[3];   // [2,E]: row0 src(paper),  row1 dst(author)
  const float* W1s_ap = (const float*)d_in[4];
  const float* W1d_ap = (const float*)d_in[5];
  const float* a1s_ap = (const float*)d_in[6];
  const float* a1d_ap = (const float*)d_in[7];
  const float* b1_ap  = (const float*)d_in[8];
  const float* W1s_pa = (const float*)d_in[9];
  const float* W1d_pa = (const float*)d_in[10];
  const float* a1s_pa = (const float*)d_in[11];
  const float* a1d_pa = (const float*)d_in[12];
  const float* b1_pa  = (const float*)d_in[13];
  const float* W2_ap  = (const float*)d_in[14];
  const float* a2s_ap = (const float*)d_in[15];
  const float* a2d_ap = (const float*)d_in[16];
  const float* b2_ap  = (const float*)d_in[17];
  const float* W2_pa  = (const float*)d_in[18];
  const float* a2s_pa = (const float*)d_in[19];
  const float* a2d_pa = (const float*)d_in[20];
  const float* b2_pa  = (const float*)d_in[21];

  float* out_all = (float*)d_out;
  float* author2 = out_all;                         // [NA,64]
  float* paper2  = out_all + (size_t)NAUT * 64;     // [NP,64]

  // workspace layout (floats): 4 big feature buffers + 4 small per-node buffers
  float* ws   = (float*)d_ws;
  const size_t BIG = (size_t)50000 * 256;
  float* hs   = ws;                 // current h_src features
  float* hd   = hs + BIG;           // current h_dst features (dead after scores)
  float* p1   = hd + BIG;           // paper1  [NP,256]
  float* a1   = p1 + BIG;           // author1 [NA,256]
  float* as_  = a1 + BIG;           // [Ns,H] scores
  float* ad_  = as_ + (size_t)50000 * 4;
  float* emax = ad_ + (size_t)50000 * 4;
  float* den  = emax + (size_t)50000 * 4;

  auto cdiv = [](int a, int b) { return (a + b - 1) / b; };

  // One full GATConv: out[Nd,H*C] = softmax-weighted aggregation + bias
  auto run_conv = [&](const float* xsrc, const float* Wsrc,
                      const float* xdst, const float* Wdst,
                      const float* atts, const float* attd, const float* bias,
                      const int* esrc, const int* edst,
                      int Ns, int Nd, int K, int H, int C, float* outbuf) {
    int HC = H * C;
    // dense projections (WMMA)
    {
      int waves = (Ns >> 4) * (HC >> 6);
      gemm_f32_wmma<<<cdiv(waves, 8), 256, 0, stream>>>(xsrc, Wsrc, hs, Ns, HC, K);
      waves = (Nd >> 4) * (HC >> 6);
      gemm_f32_wmma<<<cdiv(waves, 8), 256, 0, stream>>>(xdst, Wdst, hd, Nd, HC, K);
    }
    // attention scores
    attn_score<<<cdiv(Ns * H, 256), 256, 0, stream>>>(hs, atts, as_, Ns, H, C);
    attn_score<<<cdiv(Nd * H, 256), 256, 0, stream>>>(hd, attd, ad_, Nd, H, C);
    // init reductions + output accumulator
    fill_f32<<<cdiv(Nd * H, 256), 256, 0, stream>>>(emax, -INFINITY, Nd * H);
    hipMemsetAsync(den, 0, (size_t)Nd * H * sizeof(float), stream);
    hipMemsetAsync(outbuf, 0, (size_t)Nd * HC * sizeof(float), stream);
    // 3-pass segment softmax + scatter
    edge_logit_max<<<cdiv(NEDGE * H, 256), 256, 0, stream>>>(esrc, edst, as_, ad_, emax, NEDGE, H);
    edge_denom   <<<cdiv(NEDGE * H, 256), 256, 0, stream>>>(esrc, edst, as_, ad_, emax, den, NEDGE, H);
    edge_scatter <<<cdiv(NEDGE, 8), 256, 0, stream>>>(esrc, edst, hs, as_, ad_, emax, den,
                                                      outbuf, NEDGE, H, C);
    bias_add<<<cdiv(Nd * HC, 256), 256, 0, stream>>>(outbuf, bias, Nd * HC, HC);
  };

  // ---- layer 1 ----
  // author -> paper  (dst = paper)
  run_conv(x_author, W1s_ap, x_paper, W1d_ap, a1s_ap, a1d_ap, b1_ap,
           e_ap, e_ap + NEDGE, NAUT, NPAP, FIN, 4, 64, p1);
  // paper -> author  (dst = author)
  run_conv(x_paper, W1s_pa, x_author, W1d_pa, a1s_pa, a1d_pa, b1_pa,
           e_pa, e_pa + NEDGE, NPAP, NAUT, FIN, 4, 64, a1);

  // ---- layer 2 (heads=1, shared W per conv) ----
  // author1 -> paper2
  run_conv(a1, W2_ap, p1, W2_ap, a2s_ap, a2d_ap, b2_ap,
           e_ap, e_ap + NEDGE, NAUT, NPAP, 256, 1, 64, paper2);
  // paper1 -> author2
  run_conv(p1, W2_pa, a1, W2_pa, a2s_pa, a2d_pa, b2_pa,
           e_pa, e_pa + NEDGE, NPAP, NAUT, 256, 1, 64, author2);
}